// AFM_num_52707838657199
// MI455X (gfx1250) — compile-verified
//
#include <hip/hip_runtime.h>

// AFM forward, algebraically simplified:
//   softmax over size-1 axis == 1  =>  attention MLP is dead code.
//   sum_{i<j} x_i*x_j = 0.5*((sum_f x)^2 - sum_f x^2)   (elementwise over e)
//   out[b] = pooled[b,:] . p_w
// Bandwidth-bound: 16.8 MB read of x, 8 KB written. Final GEMV done with
// V_WMMA_F32_16X16X4_F32 (free vs. the memory stream).

typedef __attribute__((ext_vector_type(2))) float v2f;
typedef __attribute__((ext_vector_type(8))) float v8f;

#define FDIM 32
#define EDIM 64
#define NB   16          // batches per block
#define LDSS 68          // pooled row stride in floats (64 + 4 pad -> conflict-free)

__global__ __launch_bounds__(256) void afm_fused_kernel(
    const float* __restrict__ x,     // [B, F, E]
    const float* __restrict__ p_w,   // [E, 1]
    float* __restrict__ out,         // [B, 1]
    int Btotal)
{
    __shared__ float pooled[NB * LDSS];
    __shared__ float s_pw[EDIM];

    const int t = threadIdx.x;                 // 0..255
    if (t < EDIM) s_pw[t] = p_w[t];

    const int b_local = t >> 4;                // 0..15
    const int e4      = (t & 15) << 2;         // 0,4,...,60
    const int batch   = blockIdx.x * NB + b_local;

    float s0 = 0.f, s1 = 0.f, s2 = 0.f, s3 = 0.f;
    float q0 = 0.f, q1 = 0.f, q2 = 0.f, q3 = 0.f;

    if (batch < Btotal) {
        // float4 pointer to x[batch, 0, e4]
        const float4* xp = reinterpret_cast<const float4*>(x)
                         + (size_t)batch * (FDIM * EDIM / 4) + (e4 >> 2);
        #pragma unroll
        for (int f = 0; f < FDIM; ++f) {
            float4 v = xp[f * (EDIM / 4)];     // coalesced b128 stream
            s0 += v.x; q0 += v.x * v.x;
            s1 += v.y; q1 += v.y * v.y;
            s2 += v.z; q2 += v.z * v.z;
            s3 += v.w; q3 += v.w * v.w;
        }
    }

    float* pr = &pooled[b_local * LDSS + e4];
    pr[0] = 0.5f * (s0 * s0 - q0);
    pr[1] = 0.5f * (s1 * s1 - q1);
    pr[2] = 0.5f * (s2 * s2 - q2);
    pr[3] = 0.5f * (s3 * s3 - q3);

    __syncthreads();

    // Wave 0 (threads 0..31, wave-uniform branch, EXEC all ones inside):
    // OUT[16x1] = POOLED[16x64] @ p_w[64x1] via 16 chained f32 WMMAs (K=4 each).
    if (t < 32) {
        const int lane = t;
        const int m    = lane & 15;            // A-matrix row
        const int kadd = (lane >> 4) << 1;     // lanes 0-15: K0/K1, lanes 16-31: K2/K3

        v8f c = {0.f, 0.f, 0.f, 0.f, 0.f, 0.f, 0.f, 0.f};
        #pragma unroll
        for (int kc = 0; kc < 16; ++kc) {
            const int e0 = (kc << 2) + kadd;
            v2f a, b;
            a.x = pooled[m * LDSS + e0];       // A: 16x4 f32 fragment
            a.y = pooled[m * LDSS + e0 + 1];
            b.x = s_pw[e0];                    // B: p_w broadcast across all 16 columns
            b.y = s_pw[e0 + 1];
            c = __builtin_amdgcn_wmma_f32_16x16x4_f32(
                    false, a, false, b, (short)0, c, false, false);
        }
        // D layout: VGPR r, lane 0 -> (M=r, N=0); lane 16 -> (M=8+r, N=0)
        if ((lane & 15) == 0) {
            const int mbase = (lane >> 4) << 3;
            #pragma unroll
            for (int r = 0; r < 8; ++r) {
                const int ob = blockIdx.x * NB + mbase + r;
                if (ob < Btotal) out[ob] = c[r];
            }
        }
    }
}

extern "C" void kernel_launch(void* const* d_in, const int* in_sizes, int n_in,
                              void* d_out, int out_size, void* d_ws, size_t ws_size,
                              hipStream_t stream) {
    (void)n_in; (void)d_ws; (void)ws_size; (void)out_size;
    // setup_inputs order: x, W_w, W_b, h_w, h_b, p_w  (W_w/W_b/h_w/h_b are dead code)
    const float* x   = (const float*)d_in[0];
    const float* p_w = (const float*)d_in[5];
    float* out = (float*)d_out;

    const int Btotal = in_sizes[0] / (FDIM * EDIM);   // 2048
    const int grid   = (Btotal + NB - 1) / NB;        // 128 blocks of 256 threads
    afm_fused_kernel<<<grid, 256, 0, stream>>>(x, p_w, out, Btotal);
}